// CapsuleLayer_66614942761166
// MI455X (gfx1250) — compile-verified
//
#include <hip/hip_runtime.h>
#include <hip/hip_bf16.h>

// CapsuleLayer dynamic routing for MI455X (gfx1250, wave32).
//
// Roofline: W stream = 512 MiB read-once -> ~22us floor @ 23.3 TB/s HBM.
// u (32 MiB) lives in the 192 MB L2 across the 3 routing iterations.
// u_hat is mapped onto V_WMMA_F32_16X16X4_F32 via a block-diagonal B
// (batched 32x16 matvecs -> K=256 GEMM per 16-input-capsule tile).

#define NCAPS 64
#define ICAPS 4096
#define CDIM  32
#define IDIM  16
#define EPSK  1e-7f

typedef __attribute__((ext_vector_type(2))) float v2f;
typedef __attribute__((ext_vector_type(8))) float v8f;

// ---------------------------------------------------------------------------
// K0: zero logits (ws is poisoned 0xAA once; we must re-init every call)
// ---------------------------------------------------------------------------
__global__ void caps_zero_kernel(float* __restrict__ p, int nElem) {
    int idx = blockIdx.x * blockDim.x + threadIdx.x;
    if (idx < nElem) p[idx] = 0.0f;
}

// ---------------------------------------------------------------------------
// K1: u[n,i,d] = sum_k W[n,i,d,k] * x[i,k]   via WMMA f32 16x16x4.
//
// Tile = (n, 16 consecutive i, all 32 d). GEMM view with K = 16i * 16k = 256:
//   Out[d_local, i_local] = sum_kk A[d_local,kk] * B[kk,i_local]
//   A[d,(j,k)] = W[n, i0+j, dtile+d, k]          (the real 512 MiB stream)
//   B[(j,k),i] = (i==j) ? x[i0+j,k] : 0          (block-diagonal broadcast)
// Per ISA f32 A 16x4 layout: lanes 0-15 hold K=0,1 (VGPR0,1); lanes 16-31
// hold K=2,3 -> each lane loads a float2 of W at k = 4t + 2*half.
// One wave handles both d-halves (two v8f accumulators) sharing B.
// ---------------------------------------------------------------------------
__global__ __launch_bounds__(256) void caps_uhat_wmma_kernel(
        const float* __restrict__ W, const float* __restrict__ x,
        float* __restrict__ u) {
    const int wid  = threadIdx.x >> 5;
    const int lane = threadIdx.x & 31;
    const int tile = blockIdx.x * 8 + wid;      // 16384 tiles total
    const int n    = tile >> 8;                 // 256 i-groups per capsule
    const int i0   = (tile & 255) << 4;
    const int half = lane >> 4;                 // K sub-pair / M sub-block
    const int l    = lane & 15;

    const float* Wn = W + (size_t)n * ICAPS * (CDIM * IDIM);
    v8f acc0 = {};   // d = 0..15
    v8f acc1 = {};   // d = 16..31
    const v2f bz = {0.0f, 0.0f};

    for (int j = 0; j < 16; ++j) {
        const int i = i0 + j;
        const float* Wrow = Wn + (size_t)i * (CDIM * IDIM); // 32d x 16k block
        const float* xr   = x + i * IDIM;
        __builtin_prefetch(Wrow + (CDIM * IDIM), 0, 0);     // next i block
#pragma unroll
        for (int t = 0; t < 4; ++t) {
            const int k = t * 4 + half * 2;
            // A tiles (ISA 16x4 f32 layout): row d, its lane-half K pair.
            v2f a0 = *(const v2f*)(Wrow + (l)      * IDIM + k);
            v2f a1 = *(const v2f*)(Wrow + (16 + l) * IDIM + k);
            // B column j only is non-zero (block-diagonal x broadcast).
            v2f xb = *(const v2f*)(xr + k);
            v2f b  = (l == j) ? xb : bz;
            acc0 = __builtin_amdgcn_wmma_f32_16x16x4_f32(
                false, a0, false, b, (short)0, acc0, false, false);
            acc1 = __builtin_amdgcn_wmma_f32_16x16x4_f32(
                false, a1, false, b, (short)0, acc1, false, false);
        }
    }
    // D layout: VGPR r -> M = 8*half + r, N = l  => 8 contiguous d per lane.
    float* urow = u + ((size_t)n * ICAPS + (i0 + l)) * CDIM;
#pragma unroll
    for (int r = 0; r < 8; ++r) urow[half * 8 + r]      = acc0[r];
#pragma unroll
    for (int r = 0; r < 8; ++r) urow[16 + half * 8 + r] = acc1[r];
}

// ---------------------------------------------------------------------------
// K2: per capsule n: c = softmax(logits[n,:]); s[n,d] = sum_i c_i * u[n,i,d]
// One block per n (64 blocks, 256 thr = 8 waves). Deterministic tree reduce
// (wave shuffle -> LDS partials), no float atomics (graph replays bit-stable).
// ---------------------------------------------------------------------------
__global__ __launch_bounds__(256) void caps_softmax_s_kernel(
        const float* __restrict__ u, const float* __restrict__ logits,
        float* __restrict__ s) {
    const int n   = blockIdx.x;
    const int tid = threadIdx.x;
    const int wid = tid >> 5;
    const int lane = tid & 31;
    __shared__ float red[8];
    __shared__ float sPart[8][CDIM];

    const float* lrow = logits + (size_t)n * ICAPS;
    // pass 1: row max
    float m = -3.4e38f;
#pragma unroll
    for (int c = 0; c < 16; ++c) m = fmaxf(m, lrow[c * 256 + tid]);
#pragma unroll
    for (int off = 16; off; off >>= 1) m = fmaxf(m, __shfl_xor(m, off, 32));
    if (lane == 0) red[wid] = m;
    __syncthreads();
    float bmax = red[0];
#pragma unroll
    for (int w = 1; w < 8; ++w) bmax = fmaxf(bmax, red[w]);
    __syncthreads();
    // pass 2: exp + sum
    float e[16];
    float sum = 0.0f;
#pragma unroll
    for (int c = 0; c < 16; ++c) {
        e[c] = __expf(lrow[c * 256 + tid] - bmax);
        sum += e[c];
    }
#pragma unroll
    for (int off = 16; off; off >>= 1) sum += __shfl_xor(sum, off, 32);
    if (lane == 0) red[wid] = sum;
    __syncthreads();
    float tot = 0.0f;
#pragma unroll
    for (int w = 0; w < 8; ++w) tot += red[w];
    const float inv = 1.0f / tot;
    // pass 3: s accumulate (u streamed from L2)
    float sl[CDIM];
#pragma unroll
    for (int d = 0; d < CDIM; ++d) sl[d] = 0.0f;
    for (int c = 0; c < 16; ++c) {
        const float cc = e[c] * inv;
        const float* ur = u + ((size_t)n * ICAPS + (c * 256 + tid)) * CDIM;
#pragma unroll
        for (int d = 0; d < CDIM; ++d) sl[d] = fmaf(cc, ur[d], sl[d]);
    }
#pragma unroll
    for (int d = 0; d < CDIM; ++d) {
        float vs = sl[d];
#pragma unroll
        for (int off = 16; off; off >>= 1) vs += __shfl_xor(vs, off, 32);
        if (lane == 0) sPart[wid][d] = vs;
    }
    __syncthreads();
    if (tid < CDIM) {
        float acc = 0.0f;
#pragma unroll
        for (int w = 0; w < 8; ++w) acc += sPart[w][tid];
        s[n * CDIM + tid] = acc;
    }
}

// ---------------------------------------------------------------------------
// K3: faithful "global" squash: sq = sum over ALL (n,d) of s^2;
//     v = s * sq/(1+sq)/(sqrt(sq)+eps).  Single block (2048 elems).
// ---------------------------------------------------------------------------
__global__ __launch_bounds__(256) void caps_squash_kernel(
        const float* __restrict__ s, float* __restrict__ v,
        float* __restrict__ vout) {
    const int tid = threadIdx.x;
    const int wid = tid >> 5;
    const int lane = tid & 31;
    __shared__ float red[8];
    float sv[8];
    float acc = 0.0f;
#pragma unroll
    for (int c = 0; c < 8; ++c) {
        sv[c] = s[c * 256 + tid];
        acc = fmaf(sv[c], sv[c], acc);
    }
#pragma unroll
    for (int off = 16; off; off >>= 1) acc += __shfl_xor(acc, off, 32);
    if (lane == 0) red[wid] = acc;
    __syncthreads();
    float sq = 0.0f;
#pragma unroll
    for (int w = 0; w < 8; ++w) sq += red[w];
    const float scale = sq / (1.0f + sq) / (sqrtf(sq) + EPSK);
#pragma unroll
    for (int c = 0; c < 8; ++c) {
        const float val = sv[c] * scale;
        v[c * 256 + tid] = val;
        if (vout) vout[c * 256 + tid] = val;   // final iter -> d_out
    }
}

// ---------------------------------------------------------------------------
// K4: logits[n,i] += sum_d u[n,i,d] * v[n,d]   (u from L2, v via LDS)
// ---------------------------------------------------------------------------
__global__ __launch_bounds__(256) void caps_logits_update_kernel(
        const float* __restrict__ u, const float* __restrict__ v,
        float* __restrict__ logits) {
    const int n = blockIdx.x >> 4;
    const int i = ((blockIdx.x & 15) << 8) + threadIdx.x;
    __shared__ float vs[CDIM];
    if (threadIdx.x < CDIM) vs[threadIdx.x] = v[n * CDIM + threadIdx.x];
    __syncthreads();
    const float* ur = u + ((size_t)n * ICAPS + i) * CDIM;
    float dot = 0.0f;
#pragma unroll
    for (int d = 0; d < CDIM; ++d) dot = fmaf(ur[d], vs[d], dot);
    logits[(size_t)n * ICAPS + i] += dot;
}

// ---------------------------------------------------------------------------
extern "C" void kernel_launch(void* const* d_in, const int* in_sizes, int n_in,
                              void* d_out, int out_size, void* d_ws, size_t ws_size,
                              hipStream_t stream) {
    (void)in_sizes; (void)n_in; (void)out_size; (void)ws_size;
    const float* x = (const float*)d_in[0];   // (4096, 16, 1)
    const float* W = (const float*)d_in[1];   // (64, 4096, 32, 16)
    float* out = (float*)d_out;               // (64, 32, 1)

    // workspace layout (~33.1 MiB total):
    float* u      = (float*)d_ws;                         // 64*4096*32
    float* logits = u + (size_t)NCAPS * ICAPS * CDIM;     // 64*4096
    float* s      = logits + (size_t)NCAPS * ICAPS;       // 64*32
    float* v      = s + NCAPS * CDIM;                     // 64*32

    caps_zero_kernel<<<(NCAPS * ICAPS + 255) / 256, 256, 0, stream>>>(
        logits, NCAPS * ICAPS);

    // 16384 tiles (64 n x 256 i-groups), 8 waves/block -> 2048 blocks
    caps_uhat_wmma_kernel<<<2048, 256, 0, stream>>>(W, x, u);

    for (int it = 0; it < 3; ++it) {
        caps_softmax_s_kernel<<<NCAPS, 256, 0, stream>>>(u, logits, s);
        caps_squash_kernel<<<1, 256, 0, stream>>>(s, v, it == 2 ? out : nullptr);
        if (it < 2)  // reference's last logits update is dead code
            caps_logits_update_kernel<<<1024, 256, 0, stream>>>(u, v, logits);
    }
}